// Hyperpixel_SFNet_56599079026973
// MI455X (gfx1250) — compile-verified
//
#include <hip/hip_runtime.h>

#define N_PIX   250000
#define N_SUPX  50000
#define N_EDGE  800000
#define DIM     128
#define N_LAYERS 5
#define RHP     5
#define BN_EPS  1e-5f
#define SLOPE   0.01f

static constexpr float INV_2G = 1.0f / (2.0f + 0.9f);

typedef __attribute__((ext_vector_type(2))) float v2f;
typedef __attribute__((ext_vector_type(8))) float v8f;

// ---------------- fill (works for float or int buffers; zero bit-pattern) ----------------
__global__ void fill_zero(float* __restrict__ p, int n4) {
    int i = blockIdx.x * blockDim.x + threadIdx.x;
    if (i < n4) ((float4*)p)[i] = make_float4(0.f, 0.f, 0.f, 0.f);
}

// ================= CSR-by-dst build (once per launch) =================
__global__ void hist_dst(const int* __restrict__ edst, int* __restrict__ deg) {
    int e = blockIdx.x * blockDim.x + threadIdx.x;
    if (e < N_EDGE) atomicAdd(&deg[edst[e]], 1);
}

// per-block exclusive scan (256 elements) + block totals
__global__ void scan_block(const int* __restrict__ deg, int* __restrict__ rowptr,
                           int* __restrict__ bsum) {
    __shared__ int sh[256];
    int tid = threadIdx.x;
    int i = blockIdx.x * 256 + tid;
    int v = (i < N_SUPX) ? deg[i] : 0;
    sh[tid] = v;
    __syncthreads();
    for (int off = 1; off < 256; off <<= 1) {
        int t = (tid >= off) ? sh[tid - off] : 0;
        __syncthreads();
        sh[tid] += t;
        __syncthreads();
    }
    if (i < N_SUPX) rowptr[i] = sh[tid] - v;      // exclusive within block
    if (tid == 255) bsum[blockIdx.x] = sh[255];   // block total
}

// exclusive scan of block sums (NB <= 256) in one block
__global__ void scan_bsum(int* __restrict__ bsum, int nb) {
    __shared__ int sh[256];
    int tid = threadIdx.x;
    int v = (tid < nb) ? bsum[tid] : 0;
    sh[tid] = v;
    __syncthreads();
    for (int off = 1; off < 256; off <<= 1) {
        int t = (tid >= off) ? sh[tid - off] : 0;
        __syncthreads();
        sh[tid] += t;
        __syncthreads();
    }
    if (tid < nb) bsum[tid] = sh[tid] - v;        // exclusive
}

__global__ void scan_add(int* __restrict__ rowptr, const int* __restrict__ bsum) {
    int i = blockIdx.x * 256 + threadIdx.x;
    if (i < N_SUPX) rowptr[i] += bsum[blockIdx.x];
    if (i == 0) rowptr[N_SUPX] = N_EDGE;
}

__global__ void csr_scatter(const int* __restrict__ esrc, const int* __restrict__ edst,
                            const float* __restrict__ ew, const int* __restrict__ rowptr,
                            int* __restrict__ fill, int* __restrict__ csr_src,
                            float* __restrict__ csr_w) {
    int e = blockIdx.x * blockDim.x + threadIdx.x;
    if (e >= N_EDGE) return;
    int d = edst[e];
    int p = rowptr[d] + atomicAdd(&fill[d], 1);
    csr_src[p] = esrc[e];
    csr_w[p]   = ew[e];
}

// ================= pooling: segment mean =================
__global__ void pool_accum(const float* __restrict__ x, const int* __restrict__ seg,
                           float* __restrict__ hsum, float* __restrict__ counts) {
    int gid  = blockIdx.x * blockDim.x + threadIdx.x;
    int pix  = gid >> 5;
    int lane = gid & 31;
    if (pix >= N_PIX) return;
    int s = seg[pix];
    float4 v = ((const float4*)(x + (size_t)pix * DIM))[lane];
    float* o = hsum + (size_t)s * DIM + lane * 4;
    atomicAdd(o + 0, v.x); atomicAdd(o + 1, v.y);
    atomicAdd(o + 2, v.z); atomicAdd(o + 3, v.w);
    if (lane == 0) atomicAdd(&counts[s], 1.0f);
}

__global__ void pool_div(float* __restrict__ h, const float* __restrict__ counts) {
    int i = blockIdx.x * blockDim.x + threadIdx.x;   // over N_SUPX*DIM/4
    if (i >= N_SUPX * DIM / 4) return;
    int row = i / (DIM / 4);
    float c = counts[row];
    c = c > 1.0f ? c : 1.0f;
    float inv = 1.0f / c;
    float4 v = ((float4*)h)[i];
    v.x *= inv; v.y *= inv; v.z *= inv; v.w *= inv;
    ((float4*)h)[i] = v;
}

// ================= GEMM: Z = A(50000x128) @ W(128x128) + bias =================
// B (the wave's 16-col slab of W) preloaded into 64 VGPRs; inner loop = 1 A-load + 1 WMMA.
// grid 625 blocks x 8 waves; wave w of block b -> n-tile w, m-tiles {b, b+625, ..., b+2500}.
__global__ void __launch_bounds__(256) gemm_bias_wmma(
    const float* __restrict__ A, const float* __restrict__ Wm,
    const float* __restrict__ bias, float* __restrict__ Z) {
    int lane = threadIdx.x & 31;
    int nt   = threadIdx.x >> 5;        // 0..7 : n-tile
    int ws   = blockIdx.x;              // 0..624 : m-stream
    int lm = lane & 15;
    int kh = (lane >> 4) << 1;          // lanes 0-15 -> K{0,1}; 16-31 -> K{2,3}
    int n0 = nt << 4;

    // preload all 32 B fragments (whole K) into registers
    v2f bfrag[32];
    const float* bp = Wm + (size_t)kh * DIM + n0 + lm;
#pragma unroll
    for (int kk = 0; kk < 32; ++kk) {
        bfrag[kk].x = bp[(size_t)(4 * kk) * DIM];
        bfrag[kk].y = bp[(size_t)(4 * kk + 1) * DIM];
    }
    int col = n0 + lm;
    float bv = bias[col];

#pragma unroll
    for (int rep = 0; rep < 5; ++rep) {
        int mt = ws + rep * 625;        // 0..3124
        int m0 = mt << 4;
        const float* ap = A + (size_t)(m0 + lm) * DIM + kh;
        v8f acc = {0.f, 0.f, 0.f, 0.f, 0.f, 0.f, 0.f, 0.f};
#pragma unroll
        for (int kk = 0; kk < 32; ++kk) {
            v2f a = ((const v2f*)ap)[2 * kk];   // 8B-aligned global_load_b64
            acc = __builtin_amdgcn_wmma_f32_16x16x4_f32(
                false, a, false, bfrag[kk], (short)0, acc, false, false);
        }
        int r0 = m0 + ((lane >> 4) << 3);
#pragma unroll
        for (int r = 0; r < 8; ++r)
            Z[(size_t)(r0 + r) * DIM + col] = acc[r] + bv;
    }
}

// ================= CSR prop: out[r] = sum_e w_e * zin[src_e]  (+ fused combine) =================
// One wave per destination row; no atomics, no pre-zeroing.
__global__ void __launch_bounds__(256) prop_csr(
    const float* __restrict__ zin, float* __restrict__ out,
    const float* __restrict__ xstart,          // null: plain prop; else (acc+xs)*INV_2G
    const int* __restrict__ rowptr, const int* __restrict__ csr_src,
    const float* __restrict__ csr_w) {
    int wave = blockIdx.x * 8 + (threadIdx.x >> 5);
    int lane = threadIdx.x & 31;
    if (wave >= N_SUPX) return;
    int e0 = rowptr[wave], e1 = rowptr[wave + 1];
    float4 acc = make_float4(0.f, 0.f, 0.f, 0.f);
    for (int e = e0; e < e1; ++e) {
        int s   = csr_src[e];
        float w = csr_w[e];
        float4 v = ((const float4*)(zin + (size_t)s * DIM))[lane];
        acc.x = fmaf(w, v.x, acc.x); acc.y = fmaf(w, v.y, acc.y);
        acc.z = fmaf(w, v.z, acc.z); acc.w = fmaf(w, v.w, acc.w);
    }
    if (xstart) {
        float4 xs = ((const float4*)(xstart + (size_t)wave * DIM))[lane];
        acc.x = (acc.x + xs.x) * INV_2G; acc.y = (acc.y + xs.y) * INV_2G;
        acc.z = (acc.z + xs.z) * INV_2G; acc.w = (acc.w + xs.w) * INV_2G;
    }
    ((float4*)(out + (size_t)wave * DIM))[lane] = acc;
}

// ================= batch-norm =================
__global__ void bn_stats(const float* __restrict__ xi, float* __restrict__ stats) {
    int c  = threadIdx.x;                 // 0..127 : channel
    int nb = gridDim.x;
    int rows_per = (N_SUPX + nb - 1) / nb;
    int r0 = blockIdx.x * rows_per;
    int r1 = r0 + rows_per; if (r1 > N_SUPX) r1 = N_SUPX;
    float s = 0.f, s2 = 0.f;
    for (int r = r0; r < r1; ++r) {
        float v = xi[(size_t)r * DIM + c];
        s += v; s2 = fmaf(v, v, s2);
    }
    atomicAdd(&stats[c], s);
    atomicAdd(&stats[DIM + c], s2);
}

__global__ void bn_apply(const float* __restrict__ xi, const float* __restrict__ stats,
                         const float* __restrict__ gam, const float* __restrict__ bet,
                         float* __restrict__ h) {
    int i = blockIdx.x * blockDim.x + threadIdx.x;   // over N_SUPX*DIM/4
    if (i >= N_SUPX * DIM / 4) return;
    int c4 = (i % (DIM / 4)) * 4;
    const float invN = 1.0f / (float)N_SUPX;
    float4 v = ((const float4*)xi)[i];
    float4 r;
    float* rp = &r.x; const float* vp = &v.x;
#pragma unroll
    for (int j = 0; j < 4; ++j) {
        int c = c4 + j;
        float mean = stats[c] * invN;
        float var  = stats[DIM + c] * invN - mean * mean;
        float o = gam[c] * (vp[j] - mean) * rsqrtf(var + BN_EPS) + bet[c];
        rp[j] = o > 0.f ? o : SLOPE * o;
    }
    ((float4*)h)[i] = r;
}

// ================= unpool =================
__global__ void unpool(const float* __restrict__ h, const int* __restrict__ seg,
                       float* __restrict__ out) {
    int gid  = blockIdx.x * blockDim.x + threadIdx.x;
    int pix  = gid >> 5;
    int lane = gid & 31;
    if (pix >= N_PIX) return;
    int s = seg[pix];
    float4 v = ((const float4*)(h + (size_t)s * DIM))[lane];
    ((float4*)(out + (size_t)pix * DIM))[lane] = v;
}

extern "C" void kernel_launch(void* const* d_in, const int* in_sizes, int n_in,
                              void* d_out, int out_size, void* d_ws, size_t ws_size,
                              hipStream_t stream) {
    const float* x     = (const float*)d_in[0];
    const float* W     = (const float*)d_in[1];
    const float* bvec  = (const float*)d_in[2];
    const float* gamma = (const float*)d_in[3];
    const float* beta  = (const float*)d_in[4];
    const float* ew    = (const float*)d_in[5];
    const int*   seg   = (const int*)d_in[6];
    const int*   eidx  = (const int*)d_in[7];
    const int*   esrc  = eidx;
    const int*   edst  = eidx + N_EDGE;
    float* out = (float*)d_out;

    // ---- workspace carve-out (256B aligned)
    char* ws = (char*)d_ws;
    size_t off = 0;
    auto carve = [&](size_t bytes) -> void* {
        void* p = (void*)(ws + off);
        off += (bytes + 255) & ~(size_t)255;
        return p;
    };
    float* h       = (float*)carve((size_t)N_SUPX * DIM * 4);   // 25.6 MB
    float* z       = (float*)carve((size_t)N_SUPX * DIM * 4);   // gemm out / ping-pong
    float* xstart  = (float*)carve((size_t)N_SUPX * DIM * 4);
    float* xi      = (float*)carve((size_t)N_SUPX * DIM * 4);
    float* counts  = (float*)carve((size_t)N_SUPX * 4);
    float* stats   = (float*)carve(2 * DIM * 4);
    int*   deg     = (int*)carve((size_t)N_SUPX * 4);
    int*   fill    = (int*)carve((size_t)N_SUPX * 4);
    int*   rowptr  = (int*)carve((size_t)(N_SUPX + 4) * 4);
    int*   bsum    = (int*)carve(256 * 4);
    int*   csr_src = (int*)carve((size_t)N_EDGE * 4);
    float* csr_w   = (float*)carve((size_t)N_EDGE * 4);

    const int TB = 256;
    const int nHD = N_SUPX * DIM;
    const int NB_SCAN = (N_SUPX + 255) / 256;   // 196
    auto g4 = [&](int nfloats) { return (nfloats / 4 + TB - 1) / TB; };

    // ---- build CSR by destination (once per launch)
    fill_zero<<<g4(N_SUPX), TB, 0, stream>>>((float*)deg, N_SUPX / 4);
    fill_zero<<<g4(N_SUPX), TB, 0, stream>>>((float*)fill, N_SUPX / 4);
    hist_dst<<<(N_EDGE + TB - 1) / TB, TB, 0, stream>>>(edst, deg);
    scan_block<<<NB_SCAN, 256, 0, stream>>>(deg, rowptr, bsum);
    scan_bsum<<<1, 256, 0, stream>>>(bsum, NB_SCAN);
    scan_add<<<NB_SCAN, 256, 0, stream>>>(rowptr, bsum);
    csr_scatter<<<(N_EDGE + TB - 1) / TB, TB, 0, stream>>>(esrc, edst, ew, rowptr,
                                                           fill, csr_src, csr_w);

    // ---- pooling: h = segment_mean(x)
    fill_zero<<<g4(nHD), TB, 0, stream>>>(h, nHD / 4);
    fill_zero<<<g4(N_SUPX), TB, 0, stream>>>(counts, N_SUPX / 4);
    pool_accum<<<(N_PIX * 32 + TB - 1) / TB, TB, 0, stream>>>(x, seg, h, counts);
    pool_div<<<g4(nHD), TB, 0, stream>>>(h, counts);

    // ---- 5 SFNet layers
    const int PROP_GRID = N_SUPX / 8;   // 6250 blocks x 8 waves = 50000 rows
    for (int l = 0; l < N_LAYERS; ++l) {
        const float* Wl = W + (size_t)l * DIM * DIM;
        const float* bl = bvec + (size_t)l * DIM;

        // z = h @ W[l] + b[l]  (WMMA, B register-resident)
        gemm_bias_wmma<<<625, 256, 0, stream>>>(h, Wl, bl, z);

        // x_start = prop(z)   (CSR gather, no atomics)
        prop_csr<<<PROP_GRID, 256, 0, stream>>>(z, xstart, nullptr, rowptr, csr_src, csr_w);

        // xi = h; 5x: xi = (prop(xi) + x_start) / 2.9   (ping-pong h->xi->z->xi->z->xi)
        prop_csr<<<PROP_GRID, 256, 0, stream>>>(h,  xi, xstart, rowptr, csr_src, csr_w);
        prop_csr<<<PROP_GRID, 256, 0, stream>>>(xi, z,  xstart, rowptr, csr_src, csr_w);
        prop_csr<<<PROP_GRID, 256, 0, stream>>>(z,  xi, xstart, rowptr, csr_src, csr_w);
        prop_csr<<<PROP_GRID, 256, 0, stream>>>(xi, z,  xstart, rowptr, csr_src, csr_w);
        prop_csr<<<PROP_GRID, 256, 0, stream>>>(z,  xi, xstart, rowptr, csr_src, csr_w);

        // batch-norm (batch stats) + leaky relu -> h
        fill_zero<<<1, TB, 0, stream>>>(stats, (2 * DIM) / 4);
        bn_stats<<<512, DIM, 0, stream>>>(xi, stats);
        bn_apply<<<g4(nHD), TB, 0, stream>>>(xi, stats, gamma + (size_t)l * DIM,
                                             beta + (size_t)l * DIM, h);
    }

    // ---- unpool
    unpool<<<(N_PIX * 32 + TB - 1) / TB, TB, 0, stream>>>(h, seg, out);
}